// Net_16449724743713
// MI455X (gfx1250) — compile-verified
//
#include <hip/hip_runtime.h>

typedef __attribute__((ext_vector_type(2))) float v2f;
typedef __attribute__((ext_vector_type(8))) float v8f;

static __device__ __forceinline__ v8f wmma_f32_16x16x4(v2f a, v2f b, v8f c) {
  // 8-arg form: (neg_a, A, neg_b, B, c_mod, C, reuse_a, reuse_b)
  return __builtin_amdgcn_wmma_f32_16x16x4_f32(false, a, false, b, (short)0, c, false, false);
}

// ---------------- BatchNorm batch stats (per channel over B,V) ----------------
__global__ void bn_stats_kernel(const float* __restrict__ x, float* __restrict__ stats) {
  const int c = blockIdx.x;  // 8 channels
  __shared__ float ssum[256];
  __shared__ float ssq[256];
  float s = 0.f, q = 0.f;
  for (int i = threadIdx.x; i < 64 * 4096; i += 256) {
    const int b = i >> 12;
    const int v = i & 4095;
    const float val = x[(b * 8 + c) * 4096 + v];
    s += val;
    q += val * val;
  }
  ssum[threadIdx.x] = s;
  ssq[threadIdx.x] = q;
  __syncthreads();
  for (int off = 128; off > 0; off >>= 1) {
    if ((int)threadIdx.x < off) {
      ssum[threadIdx.x] += ssum[threadIdx.x + off];
      ssq[threadIdx.x] += ssq[threadIdx.x + off];
    }
    __syncthreads();
  }
  if (threadIdx.x == 0) {
    const float inv_n = 1.f / (64.f * 4096.f);
    const float mean = ssum[0] * inv_n;
    const float var = ssq[0] * inv_n - mean * mean;
    stats[c] = mean;
    stats[8 + c] = rsqrtf(var + 1e-5f);
  }
}

// normalize + transpose to x0[v][c*64 + b]  (V=4096, F=512)
__global__ void bn_apply_kernel(const float* __restrict__ x, const float* __restrict__ stats,
                                float* __restrict__ x0) {
  const int idx = blockIdx.x * blockDim.x + threadIdx.x;  // idx = v*512 + c*64 + b
  const int v = idx >> 9;
  const int c = (idx >> 6) & 7;
  const int b = idx & 63;
  x0[idx] = (x[(b * 8 + c) * 4096 + v] - stats[c]) * stats[8 + c];
}

// ---------------- fixed-degree SpMM Chebyshev step: out = alpha*L@in + beta*prev ----
// rows have exactly 32 entries at [r*32, r*32+32); block = (row r, 256-col chunk)
__global__ void spmm_cheb_kernel(const int* __restrict__ cols, const float* __restrict__ vals,
                                 const float* __restrict__ in, const float* __restrict__ prev,
                                 float* __restrict__ out, int F, float alpha, float beta) {
  __shared__ int scol[32];
  __shared__ float sval[32];
  const int r = blockIdx.x;
  const int f = blockIdx.y * blockDim.x + threadIdx.x;
  if (threadIdx.x < 32) {
    scol[threadIdx.x] = cols[r * 32 + threadIdx.x] * F;
    sval[threadIdx.x] = vals[r * 32 + threadIdx.x];
  }
  __syncthreads();
  float acc = 0.f;
#pragma unroll
  for (int j = 0; j < 32; ++j) acc = fmaf(sval[j], in[scol[j] + f], acc);
  out[r * F + f] = fmaf(beta, prev[r * F + f], alpha * acc);
}

// ---------------- dense Cheb-feature GEMM + bias + relu + maxpool4 (WMMA fp32) ----
// T: [8][VN][CIN*64] Chebyshev slabs; W: [COUT][CIN*8]; feature kk = c*8 + k.
// One wave per M-tile = (batch b, 16 consecutive v). Pool groups align with C rows.
// OUT_BCV=false: out[vp][co*64 + b]   (feeds next SpMM chain, V/4 rows)
// OUT_BCV=true : out[b*(COUT*VN/4) + co*(VN/4) + vp]  (feeds FC1, (B,16384))
template <int CIN, int COUT, int VN, bool OUT_BCV>
__global__ void conv_dense_relu_pool_kernel(const float* __restrict__ T,
                                            const float* __restrict__ W,
                                            const float* __restrict__ bias,
                                            float* __restrict__ out) {
  constexpr int F = CIN * 64;
  constexpr int KK = CIN * 8;
  constexpr int NT = COUT / 16;
  const int wave = threadIdx.x >> 5;
  const int lane = threadIdx.x & 31;
  const int tile = blockIdx.x * (blockDim.x >> 5) + wave;
  const int b = tile / (VN / 16);
  const int v0 = (tile % (VN / 16)) * 16;
  const int n = lane & 15;  // A row / B,C column within tile
  const int g = lane >> 4;  // lane half selects K pair / C row offset 8

  const float* Ap = T + (v0 + n) * F + b;  // + k*VN*F + c*64
  v8f acc[NT] = {};

  for (int kk0 = 0; kk0 < KK; kk0 += 4) {
    const int ka = kk0 + 2 * g;  // this lane-half's first K index
    const int c = ka >> 3;
    const int k = ka & 7;  // ka%8 in {0,2,4,6} -> ka and ka+1 share c
    v2f a;
    a.x = Ap[(k * VN) * F + c * 64];
    a.y = Ap[((k + 1) * VN) * F + c * 64];
#pragma unroll
    for (int t = 0; t < NT; ++t) {
      const v2f bv = *(const v2f*)(W + (t * 16 + n) * KK + ka);
      acc[t] = wmma_f32_16x16x4(a, bv, acc[t]);
    }
  }

  const int vpb = v0 >> 2;
#pragma unroll
  for (int t = 0; t < NT; ++t) {
    const int co = t * 16 + n;
    const float bs = bias[co];
    // lane holds C rows M = r + 8g, r=0..7 -> two complete pool-4 groups
    const float p0 =
        fmaxf(fmaxf(fmaxf(acc[t][0], acc[t][1]), fmaxf(acc[t][2], acc[t][3])) + bs, 0.f);
    const float p1 =
        fmaxf(fmaxf(fmaxf(acc[t][4], acc[t][5]), fmaxf(acc[t][6], acc[t][7])) + bs, 0.f);
    const int vp0 = vpb + 2 * g;
    if (OUT_BCV) {
      float* o = out + b * (COUT * (VN / 4)) + co * (VN / 4) + vp0;
      o[0] = p0;
      o[1] = p1;
    } else {
      float* o = out + vp0 * (COUT * 64) + co * 64 + b;
      o[0] = p0;
      o[COUT * 64] = p1;
    }
  }
}

// ---------------- FC1: (64,16384) @ (16384,512) + bias, sigmoid (WMMA fp32) --------
__global__ void fc1_sigmoid_kernel(const float* __restrict__ A,   // (64,16384)
                                   const float* __restrict__ Wm,  // fc1_w (512,16384)
                                   const float* __restrict__ bias,
                                   float* __restrict__ out) {  // (64,512)
  const int wave = threadIdx.x >> 5;
  const int lane = threadIdx.x & 31;
  const int tile = blockIdx.x * (blockDim.x >> 5) + wave;  // 0..127
  const int mt = tile >> 5;                                // 4 M-tiles
  const int nt = tile & 31;                                // 32 N-tiles
  const int n = lane & 15;
  const int g = lane >> 4;
  const float* arow = A + (mt * 16 + n) * 16384;
  const float* brow = Wm + (nt * 16 + n) * 16384;
  v8f acc = {};
  for (int kk0 = 0; kk0 < 16384; kk0 += 4) {
    const int ka = kk0 + 2 * g;
    const v2f a = *(const v2f*)(arow + ka);
    const v2f bv = *(const v2f*)(brow + ka);
    acc = wmma_f32_16x16x4(a, bv, acc);
  }
  const float bs = bias[nt * 16 + n];
#pragma unroll
  for (int r = 0; r < 8; ++r) {
    const int m = mt * 16 + r + 8 * g;
    const float v = acc[r] + bs;
    out[m * 512 + nt * 16 + n] = 1.f / (1.f + expf(-v));
  }
}

// ---------------- FC2: (64,512) @ (512,2) + bias ----------------
__global__ void fc2_kernel(const float* __restrict__ h, const float* __restrict__ w,
                           const float* __restrict__ bias, float* __restrict__ out) {
  const int t = threadIdx.x;  // 128 threads -> 64x2 outputs
  const int b = t >> 1, j = t & 1;
  float acc = bias[j];
  for (int i = 0; i < 512; ++i) acc = fmaf(h[b * 512 + i], w[j * 512 + i], acc);
  out[b * 2 + j] = acc;
}

extern "C" void kernel_launch(void* const* d_in, const int* in_sizes, int n_in,
                              void* d_out, int out_size, void* d_ws, size_t ws_size,
                              hipStream_t stream) {
  (void)in_sizes; (void)n_in; (void)out_size; (void)ws_size;
  const float* x = (const float*)d_in[0];
  const int* cols1 = (const int*)d_in[2];
  const float* vals1 = (const float*)d_in[3];
  const int* cols2 = (const int*)d_in[5];
  const float* vals2 = (const float*)d_in[6];
  const float* w1 = (const float*)d_in[7];
  const float* b1 = (const float*)d_in[8];
  const float* w2 = (const float*)d_in[9];
  const float* b2 = (const float*)d_in[10];
  const float* fc1w = (const float*)d_in[11];
  const float* fc1b = (const float*)d_in[12];
  const float* fc2w = (const float*)d_in[13];
  const float* fc2b = (const float*)d_in[14];
  float* out = (float*)d_out;

  float* ws = (float*)d_ws;
  float* stats = ws;                 // 64 floats
  float* T1 = ws + 64;               // 8 slabs of 4096*512
  const int SLAB1 = 4096 * 512;
  float* T2 = T1 + 8 * SLAB1;        // 8 slabs of 1024*2048 (slab0 = conv1 out)
  const int SLAB2 = 1024 * 2048;
  float* fc_in = T1;                 // reuse: 64*16384 floats
  float* fc1_out = T1 + 64 * 16384;  // 64*512 floats

  bn_stats_kernel<<<8, 256, 0, stream>>>(x, stats);
  bn_apply_kernel<<<(4096 * 512) / 256, 256, 0, stream>>>(x, stats, T1);

  // Chebyshev chain on graph 1 (V=4096, F=512): t1=L@t0; tk=2*L@t(k-1)-t(k-2)
  for (int k = 1; k < 8; ++k) {
    const float alpha = (k == 1) ? 1.f : 2.f;
    const float beta = (k == 1) ? 0.f : -1.f;
    const float* in = T1 + (k - 1) * SLAB1;
    const float* prev = (k == 1) ? in : T1 + (k - 2) * SLAB1;
    spmm_cheb_kernel<<<dim3(4096, 2), 256, 0, stream>>>(cols1, vals1, in, prev,
                                                        T1 + k * SLAB1, 512, alpha, beta);
  }
  // dense conv1 + relu + pool4 -> T2[0] in (V2=1024, 32*64) layout
  conv_dense_relu_pool_kernel<8, 32, 4096, false>
      <<<4096, 128, 0, stream>>>(T1, w1, b1, T2);

  // Chebyshev chain on graph 2 (V=1024, F=2048)
  for (int k = 1; k < 8; ++k) {
    const float alpha = (k == 1) ? 1.f : 2.f;
    const float beta = (k == 1) ? 0.f : -1.f;
    const float* in = T2 + (k - 1) * SLAB2;
    const float* prev = (k == 1) ? in : T2 + (k - 2) * SLAB2;
    spmm_cheb_kernel<<<dim3(1024, 8), 256, 0, stream>>>(cols2, vals2, in, prev,
                                                        T2 + k * SLAB2, 2048, alpha, beta);
  }
  // dense conv2 + relu + pool4 -> fc_in in (B, 16384) layout
  conv_dense_relu_pool_kernel<32, 64, 1024, true>
      <<<1024, 128, 0, stream>>>(T2, w2, b2, fc_in);

  fc1_sigmoid_kernel<<<32, 128, 0, stream>>>(fc_in, fc1w, fc1b, fc1_out);
  fc2_kernel<<<1, 128, 0, stream>>>(fc1_out, fc2w, fc2b, out);
}